// GCNLayer_19911468384503
// MI455X (gfx1250) — compile-verified
//
#include <hip/hip_runtime.h>

typedef float v2f __attribute__((ext_vector_type(2)));
typedef float v8f __attribute__((ext_vector_type(8)));

#define D 512
#define LDS_STRIDE 516  // 512 + 4 floats padding -> lane m hits bank 4m%64: conflict-free per 16-lane group
#define ROWS 32         // block row-tile: each wave computes 2 stacked 16x16 tiles sharing B

// ---------------------------------------------------------------------------
// Kernel 1: hw[r][c] = (sum_k h[r][k] * W[k][c]) * norm[r]
// Block: 256 threads (8 waves). Block tile: 32 rows x 128 cols.
// Each wave: two 16x16 tiles (rows mn and mn+16) via V_WMMA_F32_16X16X4_F32,
// sharing one B fragment per k-step (halves W L2 traffic, hides WMMA RAW).
// ---------------------------------------------------------------------------
__global__ __launch_bounds__(256)
void gcn_gemm_norm(const float* __restrict__ h,
                   const float* __restrict__ w,
                   const float* __restrict__ norm,
                   float* __restrict__ hw_out,
                   int N)
{
    __shared__ float ldsA[ROWS * LDS_STRIDE];
    __shared__ float normS[ROWS];

    const int rowBase = blockIdx.x * ROWS;
    const int wave    = threadIdx.x >> 5;
    const int lane    = threadIdx.x & 31;
    const int colBase = blockIdx.y * 128 + wave * 16;

    // Cooperative load of the 32x512 A strip into LDS (coalesced rows).
    for (int i = threadIdx.x; i < ROWS * D; i += 256) {
        const int r = i >> 9;           // 0..31
        const int c = i & (D - 1);      // 0..511
        int gr = rowBase + r;
        if (gr >= N) gr = N - 1;        // clamp (tail block only)
        ldsA[r * LDS_STRIDE + c] = h[gr * D + c];
    }
    if (threadIdx.x < ROWS) {
        int gr = rowBase + threadIdx.x;
        if (gr >= N) gr = N - 1;
        normS[threadIdx.x] = norm[gr];
    }
    __syncthreads();

    // Fragment geometry (ISA 7.12.2, wave32):
    //  A 16x4 f32: lanes 0-15 -> M=lane, VGPRs K{0,1}; lanes 16-31 -> K{2,3}
    //  B 4x16 f32: lanes 0-15 -> N=lane, VGPRs K{0,1}; lanes 16-31 -> K{2,3}
    const int mn   = lane & 15;
    const int kOff = (lane >> 4) << 1;     // 0 or 2

    v8f acc0 = {};   // rows rowBase+0..15
    v8f acc1 = {};   // rows rowBase+16..31
    #pragma unroll 4
    for (int k0 = 0; k0 < D; k0 += 4) {
        const int ka = k0 + kOff;
        v2f a0, a1, b;
        b.x  = w[ka * D + colBase + mn];          // 64B coalesced per half-wave, W is L2-resident
        b.y  = w[(ka + 1) * D + colBase + mn];
        a0.x = ldsA[mn * LDS_STRIDE + ka];
        a0.y = ldsA[mn * LDS_STRIDE + ka + 1];
        a1.x = ldsA[(mn + 16) * LDS_STRIDE + ka];
        a1.y = ldsA[(mn + 16) * LDS_STRIDE + ka + 1];
        // (neg_a, A, neg_b, B, c_mod, C, reuse_a, reuse_b)
        acc0 = __builtin_amdgcn_wmma_f32_16x16x4_f32(false, a0, false, b,
                                                     (short)0, acc0, false, false);
        acc1 = __builtin_amdgcn_wmma_f32_16x16x4_f32(false, a1, false, b,
                                                     (short)0, acc1, false, false);
    }

    // C/D layout: VGPR j, lanes 0-15 -> M=j ; lanes 16-31 -> M=j+8 ; N = lane&15
    const int hi = lane >> 4;
    if (rowBase + ROWS <= N) {
        // Full tile: branch-free, 16 coalesced stores per wave.
        #pragma unroll
        for (int j = 0; j < 8; ++j) {
            const int r0 = j + 8 * hi;
            hw_out[(rowBase + r0) * D + colBase + mn]      = acc0[j] * normS[r0];
            hw_out[(rowBase + 16 + r0) * D + colBase + mn] = acc1[j] * normS[16 + r0];
        }
    } else {
        // Tail block: guarded.
        #pragma unroll
        for (int j = 0; j < 8; ++j) {
            const int r0 = j + 8 * hi;
            if (rowBase + r0 < N)
                hw_out[(rowBase + r0) * D + colBase + mn] = acc0[j] * normS[r0];
            if (rowBase + 16 + r0 < N)
                hw_out[(rowBase + 16 + r0) * D + colBase + mn] = acc1[j] * normS[16 + r0];
        }
    }
}

// ---------------------------------------------------------------------------
// Kernel 0: zero the accumulator (d_out)
// ---------------------------------------------------------------------------
__global__ __launch_bounds__(256)
void gcn_zero(float* __restrict__ p, int n)
{
    const int i = blockIdx.x * 256 + threadIdx.x;
    if (i < n) p[i] = 0.0f;
}

// ---------------------------------------------------------------------------
// Kernel 2: agg[dst[e]][:] += hw[src[e]][:]
// One 128-thread block per edge; each lane: float4 read + 4 f32 atomics.
// hw (20 MB) and agg (20 MB) both fit in the 192 MB L2 -> atomics run at L2.
// ---------------------------------------------------------------------------
__global__ __launch_bounds__(128)
void gcn_scatter(const float* __restrict__ hw,
                 const int* __restrict__ src,
                 const int* __restrict__ dst,
                 float* __restrict__ agg)
{
    const int e = blockIdx.x;
    const int s = src[e];
    const int d = dst[e];
    const int f = threadIdx.x << 2;          // 0..508 step 4

    const float4 v = *reinterpret_cast<const float4*>(hw + s * D + f);
    float* p = agg + d * D + f;
    atomicAdd(p + 0, v.x);
    atomicAdd(p + 1, v.y);
    atomicAdd(p + 2, v.z);
    atomicAdd(p + 3, v.w);
}

// ---------------------------------------------------------------------------
// Kernel 3: out = agg * norm[row] + bias[col]   (in place on d_out)
// ---------------------------------------------------------------------------
__global__ __launch_bounds__(256)
void gcn_finalize(float* __restrict__ out,
                  const float* __restrict__ norm,
                  const float* __restrict__ bias,
                  int total)
{
    const int i = blockIdx.x * 256 + threadIdx.x;
    if (i < total) {
        const int row = i >> 9;          // / 512
        const int col = i & (D - 1);     // % 512
        out[i] = out[i] * norm[row] + bias[col];
    }
}

// ---------------------------------------------------------------------------
// Host-side launcher
// inputs: 0:h [N,512] f32, 1:norm [N,1] f32, 2:weight [512,512] f32,
//         3:bias [512] f32, 4:src [E] i32, 5:dst [E] i32
// d_out: [N,512] f32.  d_ws: hw scratch [N,512] f32.
// ---------------------------------------------------------------------------
extern "C" void kernel_launch(void* const* d_in, const int* in_sizes, int n_in,
                              void* d_out, int out_size, void* d_ws, size_t ws_size,
                              hipStream_t stream)
{
    const float* h    = (const float*)d_in[0];
    const float* norm = (const float*)d_in[1];
    const float* w    = (const float*)d_in[2];
    const float* bias = (const float*)d_in[3];
    const int*   src  = (const int*)d_in[4];
    const int*   dst  = (const int*)d_in[5];

    const int N = in_sizes[0] / D;       // 10000
    const int E = in_sizes[4];           // 160000
    const int total = N * D;

    float* hw  = (float*)d_ws;           // N*D floats = 20.5 MB
    float* out = (float*)d_out;          // doubles as the atomic accumulator

    // 1) zero accumulator
    gcn_zero<<<(total + 255) / 256, 256, 0, stream>>>(out, total);

    // 2) hw = (h @ W) * norm   — fp32 WMMA, 32x128 block tiles
    dim3 grid((N + ROWS - 1) / ROWS, D / 128);
    gcn_gemm_norm<<<grid, 256, 0, stream>>>(h, w, norm, hw, N);

    // 3) scatter-add over edges
    gcn_scatter<<<E, 128, 0, stream>>>(hw, src, dst, out);

    // 4) out = out * norm + bias
    gcn_finalize<<<(total + 255) / 256, 256, 0, stream>>>(out, norm, bias, total);
}